// SelfSimilarity_90494960927390
// MI455X (gfx1250) — compile-verified
//
#include <hip/hip_runtime.h>

typedef float v2f __attribute__((ext_vector_type(2)));
typedef float v8f __attribute__((ext_vector_type(8)));

constexpr int   BATCH = 8;
constexpr int   N     = 2048;
constexpr int   D     = 512;
constexpr float INV_T = 1.0f / 13.544f;

// ---------------------------------------------------------------------------
// Kernel 1: row squared-norms. One wave32 per row; float4 loads + xor-shuffle.
// ---------------------------------------------------------------------------
__global__ __launch_bounds__(256) void SelfSim_sqnorm_kernel(
    const float* __restrict__ x, float* __restrict__ sqn) {
  const int row  = blockIdx.x * 8 + (threadIdx.x >> 5);
  const int lane = threadIdx.x & 31;
  const float4* xr = (const float4*)(x + (size_t)row * D);
  float acc = 0.0f;
#pragma unroll
  for (int i = 0; i < (D / 4) / 32; ++i) {       // 4 iters: 32 lanes * float4
    float4 v = xr[lane + i * 32];
    acc += v.x * v.x + v.y * v.y + v.z * v.z + v.w * v.w;
  }
  acc += __shfl_xor(acc, 16, 32);
  acc += __shfl_xor(acc,  8, 32);
  acc += __shfl_xor(acc,  4, 32);
  acc += __shfl_xor(acc,  2, 32);
  acc += __shfl_xor(acc,  1, 32);
  if (lane == 0) sqn[row] = acc;
}

// ---------------------------------------------------------------------------
// Kernel 2: Gram via v_wmma_f32_16x16x4_f32 with 2x2 register blocking,
// fused exp(-dist/T) + row sums.
// Grid: BATCH * (N/32) blocks of 256 threads (8 waves).
// Block owns rows [i0, i0+32); each wave owns 2 i-tiles x 2 j-tiles, so each
// A and B fragment feeds two WMMAs (0.5 global loads per WMMA).
// ---------------------------------------------------------------------------
__global__ __launch_bounds__(256) void SelfSim_gram_exp_kernel(
    const float* __restrict__ x, const float* __restrict__ sqn,
    float* __restrict__ rowsum, float* __restrict__ out) {
  __shared__ float Atile[32 * D];   // 64 KB contiguous A tile (32 rows)
  __shared__ float sred[8 * 32];    // per-wave row partial sums
  __shared__ float sqi_sh[32];

  const int bt = blockIdx.x;
  const int b  = bt >> 6;           // / (N/32)
  const int i0 = (bt & 63) << 5;
  const float* xb  = x   + (size_t)b * N * D;
  const float* sqb = sqn + b * N;

  const int tid = threadIdx.x;
  {  // cooperative A-tile copy: 32 rows are contiguous 64 KB in memory
    const float4* src = (const float4*)(xb + (size_t)i0 * D);
    float4* dst = (float4*)Atile;
#pragma unroll
    for (int v = 0; v < (32 * D / 4) / 256; ++v)   // 16 float4 per thread
      dst[tid + v * 256] = src[tid + v * 256];
  }
  if (tid < 32) sqi_sh[tid] = sqb[i0 + tid];
  __syncthreads();

  const int wave  = tid >> 5;
  const int lane  = tid & 31;
  const int lrow  = lane & 15;      // M (A/C row) or N (B/C col) within tile
  const int khalf = lane >> 4;      // selects K pair {0,1} vs {2,3}

  const float* Ap0 = &Atile[lrow * D + 2 * khalf];
  const float* Ap1 = Ap0 + 16 * D;

  v8f sum0 = {};                    // rows i0..i0+15 partial sums
  v8f sum1 = {};                    // rows i0+16..i0+31 partial sums

  for (int pj = wave; pj < N / 32; pj += 8) {     // 8 uniform iters per wave
    const int j0 = pj * 32;
    v8f c00 = {};
    v8f c01 = {};
    v8f c10 = {};
    v8f c11 = {};
    const float* Bp0 = xb + (size_t)(j0 + lrow) * D + 2 * khalf;
    const float* Bp1 = Bp0 + 16 * D;
#pragma unroll 4
    for (int kb = 0; kb < D; kb += 4) {
      v2f a0 = *(const v2f*)(Ap0 + kb);   // ds_load_b64 (pairable 2addr)
      v2f a1 = *(const v2f*)(Ap1 + kb);
      v2f b0 = *(const v2f*)(Bp0 + kb);   // global_load_b64 (L2-resident)
      v2f b1 = *(const v2f*)(Bp1 + kb);
      c00 = __builtin_amdgcn_wmma_f32_16x16x4_f32(false, a0, false, b0,
                                                  (short)0, c00, false, false);
      c01 = __builtin_amdgcn_wmma_f32_16x16x4_f32(false, a0, false, b1,
                                                  (short)0, c01, false, false);
      c10 = __builtin_amdgcn_wmma_f32_16x16x4_f32(false, a1, false, b0,
                                                  (short)0, c10, false, false);
      c11 = __builtin_amdgcn_wmma_f32_16x16x4_f32(false, a1, false, b1,
                                                  (short)0, c11, false, false);
    }
    const float sqj0 = sqb[j0 + lrow];
    const float sqj1 = sqb[j0 + 16 + lrow];
    float* orow0 = out + ((size_t)(b * N + i0 + khalf * 8) * N) + j0 + lrow;
    float* orow1 = orow0 + (size_t)16 * N;
#pragma unroll
    for (int r = 0; r < 8; ++r) {         // C: VGPR r -> row r + 8*khalf
      const float sqi0 = sqi_sh[r + 8 * khalf];
      const float sqi1 = sqi_sh[16 + r + 8 * khalf];
      // row max of scores is 0 (diagonal, dist=0) -> single-pass exp is safe
      float e00 = __expf(-fmaxf(sqi0 + sqj0 - 2.0f * c00[r], 0.0f) * INV_T);
      float e01 = __expf(-fmaxf(sqi0 + sqj1 - 2.0f * c01[r], 0.0f) * INV_T);
      float e10 = __expf(-fmaxf(sqi1 + sqj0 - 2.0f * c10[r], 0.0f) * INV_T);
      float e11 = __expf(-fmaxf(sqi1 + sqj1 - 2.0f * c11[r], 0.0f) * INV_T);
      sum0[r] += e00 + e01;
      sum1[r] += e10 + e11;
      orow0[(size_t)r * N]      = e00;
      orow0[(size_t)r * N + 16] = e01;
      orow1[(size_t)r * N]      = e10;
      orow1[(size_t)r * N + 16] = e11;
    }
  }

  // reduce row sums across the 16 lanes of each half-wave
#pragma unroll
  for (int r = 0; r < 8; ++r) {
    float s0 = sum0[r];
    float s1 = sum1[r];
    s0 += __shfl_xor(s0, 1, 32);
    s0 += __shfl_xor(s0, 2, 32);
    s0 += __shfl_xor(s0, 4, 32);
    s0 += __shfl_xor(s0, 8, 32);
    s1 += __shfl_xor(s1, 1, 32);
    s1 += __shfl_xor(s1, 2, 32);
    s1 += __shfl_xor(s1, 4, 32);
    s1 += __shfl_xor(s1, 8, 32);
    if (lrow == 0) {
      sred[wave * 32 + khalf * 8 + r]      = s0;
      sred[wave * 32 + 16 + khalf * 8 + r] = s1;
    }
  }
  __syncthreads();
  if (tid < 32) {
    float t = 0.0f;
#pragma unroll
    for (int w = 0; w < 8; ++w) t += sred[w * 32 + tid];
    rowsum[b * N + i0 + tid] = t;
  }
}

// ---------------------------------------------------------------------------
// Kernel 3: normalize (bandwidth-bound; output is L2-resident at 134 MB)
// ---------------------------------------------------------------------------
__global__ __launch_bounds__(256) void SelfSim_normalize_kernel(
    float* __restrict__ out, const float* __restrict__ rowsum) {
  const size_t i4  = (size_t)blockIdx.x * 256 + threadIdx.x;
  const size_t row = (i4 * 4) >> 11;          // / N
  const float inv  = 1.0f / rowsum[row];
  float4 v = ((float4*)out)[i4];
  v.x *= inv; v.y *= inv; v.z *= inv; v.w *= inv;
  ((float4*)out)[i4] = v;
}

// ---------------------------------------------------------------------------
extern "C" void kernel_launch(void* const* d_in, const int* in_sizes, int n_in,
                              void* d_out, int out_size, void* d_ws, size_t ws_size,
                              hipStream_t stream) {
  (void)in_sizes; (void)n_in; (void)out_size; (void)ws_size;
  const float* x   = (const float*)d_in[0];
  float* out       = (float*)d_out;
  float* sqn       = (float*)d_ws;            // BATCH*N floats (64 KB)
  float* rowsum    = sqn + BATCH * N;         // BATCH*N floats (64 KB)

  SelfSim_sqnorm_kernel<<<BATCH * N / 8, 256, 0, stream>>>(x, sqn);
  SelfSim_gram_exp_kernel<<<BATCH * (N / 32), 256, 0, stream>>>(x, sqn, rowsum, out);
  SelfSim_normalize_kernel<<<((size_t)BATCH * N * N / 4) / 256, 256, 0, stream>>>(out, rowsum);
}